// Kuramoto_14714557956640
// MI455X (gfx1250) — compile-verified
//
#include <hip/hip_runtime.h>
#include <hip/hip_bf16.h>
#include <math.h>

// Problem dims (fixed by reference)
#define T_STEPS 64
#define BATCH   128
#define NIN     28
#define NH      512
#define NOUT    10
#define TWO_PI_F 6.2831853071795864769f
#define INV_TWO_PI_F 0.15915494309189533577f

typedef float v2f __attribute__((ext_vector_type(2)));
typedef float v4f __attribute__((ext_vector_type(4)));
typedef float v8f __attribute__((ext_vector_type(8)));

// Workspace layout (in floats). Total = 524288 floats = 2 MB.
//   WhT2 : element (k, i) at ((k>>1)*NH + i)*2 + (k&1)          (= Wh[i][k])
//   scP  : per (k-pair k2, batch b) a 16-byte group at (k2*BATCH + b)*4:
//          [ sin(s[b][2k2]), sin(s[b][2k2+1]), cos(s[b][2k2]), cos(s[b][2k2+1]) ]
//          -> ONE global_load_b128 yields both WMMA A operands (sin pair, cos pair)
#define OFF_WHT   0                          // 512*512
#define OFF_SC    (NH*NH)                    // 512/2 * 128 * 4 = 131072
#define OFF_STATE (OFF_SC + NH*BATCH*2)      // state[b*NH + h]  (128*512)
#define OFF_INP   (OFF_STATE + BATCH*NH)     // inp_t[b*NH + h]  (128*512)

// ---------------------------------------------------------------------------
// One-time: transpose+pack Wh so WMMA B-operands are contiguous v2f loads.
__global__ void k_transpose_wh(const float* __restrict__ Wh, float* __restrict__ WhT2) {
    int idx = blockIdx.x * blockDim.x + threadIdx.x;   // idx = j*NH + i
    int j = idx >> 9;          // k index
    int i = idx & (NH - 1);    // n index
    WhT2[((j >> 1) * NH + i) * 2 + (j & 1)] = Wh[i * NH + j];
}

__global__ void k_init_state(float* __restrict__ state) {
    int idx = blockIdx.x * blockDim.x + threadIdx.x;
    state[idx] = 0.0f;
}

// ---------------------------------------------------------------------------
// Per-step prep: sincos of current state (sin/cos-interleaved, K-pair-major
// layout for single-b128 WMMA A-loads) + input projection
// inp_t[b,h] = x[t,b,:] . Wi_w[h,:] + Wi_b[h].
__global__ void __launch_bounds__(256) k_step_prep(
    const float* __restrict__ x, const float* __restrict__ Wi_w,
    const float* __restrict__ Wi_b, float* __restrict__ ws, int t)
{
    float* __restrict__ scP = ws + OFF_SC;
    const float* __restrict__ state = ws + OFF_STATE;
    float* __restrict__ inp_t = ws + OFF_INP;

    int idx = blockIdx.x * blockDim.x + threadIdx.x;   // idx = b*NH + h
    int b = idx >> 9;
    int h = idx & (NH - 1);

    float s = state[idx];
    int base = ((h >> 1) * BATCH + b) * 4 + (h & 1);
    scP[base]     = sinf(s);
    scP[base + 2] = cosf(s);

    const float* xr = x + ((size_t)t * BATCH + b) * NIN;
    const float* wr = Wi_w + h * NIN;
    float acc = Wi_b[h];
#pragma unroll
    for (int i = 0; i < NIN; ++i) acc = fmaf(xr[i], wr[i], acc);
    inp_t[idx] = acc;
}

// ---------------------------------------------------------------------------
// Per-step WMMA kernel: C = cosS @ WhT, S = sinS @ WhT via V_WMMA_F32_16X16X4_F32.
// B tile (shared by all 8 waves of the block) staged in LDS (ds_load_b64);
// A operands come as one b128 load each iteration (sin pair + cos pair).
// Then fused update: s' = mod(sin(s)*C - cos(s)*S + inp + omega + s, 2pi).
// Grid: 32 blocks (n-tiles) x 8 waves (m-tiles) = all 256 16x16 output tiles.
__global__ void __launch_bounds__(256) k_step_wmma(
    float* __restrict__ ws, const float* __restrict__ omega)
{
    const v2f*   __restrict__ WhT2  = (const v2f*)(ws + OFF_WHT);
    const v4f*   __restrict__ scP4  = (const v4f*)(ws + OFF_SC);
    const float* __restrict__ scP   = ws + OFF_SC;
    float*       __restrict__ state = ws + OFF_STATE;
    const float* __restrict__ inp   = ws + OFF_INP;

    const int lane  = threadIdx.x & 31;
    const int wv    = threadIdx.x >> 5;
    const int l15   = lane & 15;
    const int hi    = lane >> 4;             // 0: lanes 0-15, 1: lanes 16-31
    const int mbase = wv * 16;               // batch tile (per wave)
    const int nbase = blockIdx.x * 16;       // hidden tile (per block)
    const int m = mbase + l15;
    const int n = nbase + l15;

    // Stage this block's B tile: all K=512 x 16 n-columns, pair-interleaved.
    // 4096 v2f = 32 KB of LDS; identical data is consumed by all 8 waves.
    __shared__ v2f lds_b[(NH / 2) * 16];
    for (int p = threadIdx.x; p < (NH / 2) * 16; p += 256) {
        int k2   = p >> 4;        // k-pair index
        int ncol = p & 15;
        lds_b[p] = WhT2[k2 * NH + nbase + ncol];
    }
    __syncthreads();

    v8f acc_c = {};
    v8f acc_s = {};

    // ISA A layout (16x4 f32): v0 = K=0 (lanes 0-15) / K=2 (lanes 16-31); v1 = K=1/K=3.
    // ISA B layout (4x16 f32): v0 = row K=0 / K=2; v1 = row K=1 / K=3.
    for (int k0 = 0; k0 < NH; k0 += 4) {
        const int k2 = (k0 >> 1) + hi;       // pair index of this half-wave's kA
        v4f sc = scP4[k2 * BATCH + m];       // one global_load_b128: sin+cos pairs
        v2f a_s; a_s.x = sc.x; a_s.y = sc.y;
        v2f a_c; a_c.x = sc.z; a_c.y = sc.w;
        v2f bm  = lds_b[k2 * 16 + l15];      // ds_load_b64, conflict-free
        acc_c = __builtin_amdgcn_wmma_f32_16x16x4_f32(
            false, a_c, false, bm, (short)0, acc_c, false, false);
        acc_s = __builtin_amdgcn_wmma_f32_16x16x4_f32(
            false, a_s, false, bm, (short)0, acc_s, false, false);
    }

    // C/D layout: VGPR r -> M = r (lanes 0-15) or r+8 (lanes 16-31); N = lane&15.
#pragma unroll
    for (int r = 0; r < 8; ++r) {
        const int mm = mbase + r + hi * 8;
        const int nn = n;
        const int pk = ((nn >> 1) * BATCH + mm) * 4 + (nn & 1);
        const float sn = scP[pk];
        const float cs = scP[pk + 2];
        const float coupling = sn * acc_c[r] - cs * acc_s[r];
        float v = coupling + inp[mm * NH + nn] + omega[nn] + state[mm * NH + nn];
        v -= TWO_PI_F * floorf(v * INV_TWO_PI_F);   // jnp.remainder semantics
        state[mm * NH + nn] = v;                    // in-place safe: 1 owner lane
    }
}

// ---------------------------------------------------------------------------
// Readout: out[b,o] = state[b,:] . readout_w[o,:] + readout_b[o]
__global__ void k_readout(const float* __restrict__ state,
                          const float* __restrict__ rw,
                          const float* __restrict__ rb,
                          float* __restrict__ out)
{
    int idx = blockIdx.x * blockDim.x + threadIdx.x;
    if (idx >= BATCH * NOUT) return;
    int b = idx / NOUT;
    int o = idx - b * NOUT;
    const float* sr = state + b * NH;
    const float* wr = rw + o * NH;
    float acc = rb[o];
    for (int h = 0; h < NH; ++h) acc = fmaf(sr[h], wr[h], acc);
    out[idx] = acc;
}

// ---------------------------------------------------------------------------
extern "C" void kernel_launch(void* const* d_in, const int* in_sizes, int n_in,
                              void* d_out, int out_size, void* d_ws, size_t ws_size,
                              hipStream_t stream) {
    const float* x     = (const float*)d_in[0];
    const float* Wi_w  = (const float*)d_in[1];
    const float* Wi_b  = (const float*)d_in[2];
    const float* Wh    = (const float*)d_in[3];
    const float* omega = (const float*)d_in[4];
    const float* rw    = (const float*)d_in[5];
    const float* rb    = (const float*)d_in[6];
    float* ws  = (float*)d_ws;
    float* out = (float*)d_out;

    k_transpose_wh<<<(NH * NH) / 256, 256, 0, stream>>>(Wh, ws + OFF_WHT);
    k_init_state<<<(BATCH * NH) / 256, 256, 0, stream>>>(ws + OFF_STATE);

    for (int t = 0; t < T_STEPS; ++t) {
        k_step_prep<<<(BATCH * NH) / 256, 256, 0, stream>>>(x, Wi_w, Wi_b, ws, t);
        k_step_wmma<<<NH / 16, 256, 0, stream>>>(ws, omega);
    }

    k_readout<<<(BATCH * NOUT + 255) / 256, 256, 0, stream>>>(
        ws + OFF_STATE, rw, rb, out);
}